// RouteNetModel_59803124629709
// MI455X (gfx1250) — compile-verified
//
#include <hip/hip_runtime.h>
#include <math.h>

// ---------------------------------------------------------------------------
// RouteNet on MI455X (gfx1250, wave32), f32 WMMA 16x16x4 everywhere.
// ---------------------------------------------------------------------------

typedef __attribute__((ext_vector_type(2))) float v2f;
typedef __attribute__((ext_vector_type(8))) float v8f;

#define L_DIM 32
#define P_DIM 32
#define GATE  96      // 3 * 32
#define RO    256
#define T_ITERS 8
#define MAX_LEN 15

static __device__ __forceinline__ v8f wmma4(v2f a, v2f b, v8f c) {
  // V_WMMA_F32_16X16X4_F32 : D[16x16] = A[16x4] * B[4x16] + C
  return __builtin_amdgcn_wmma_f32_16x16x4_f32(
      /*neg_a=*/false, a, /*neg_b=*/false, b,
      /*c_mod=*/(short)0, c, /*reuse_a=*/false, /*reuse_b=*/false);
}

static __device__ __forceinline__ float sigmoidf_(float x) {
  return 1.0f / (1.0f + expf(-x));
}

// ---------------------------------------------------------------------------
// Hop table construction: hop[p*15+s] = link id, plen[p] = #hops
// ---------------------------------------------------------------------------
__global__ void rn_init_hops(int* hop, int* plen, int n_paths) {
  int i = blockIdx.x * blockDim.x + threadIdx.x;
  if (i < n_paths * MAX_LEN) hop[i] = -1;
  if (i < n_paths)           plen[i] = 0;
}

__global__ void rn_fill_hops(const int* __restrict__ path_ids,
                             const int* __restrict__ seq,
                             const int* __restrict__ l2p,
                             int* hop, int* plen, int E) {
  int e = blockIdx.x * blockDim.x + threadIdx.x;
  if (e >= E) return;
  int p = path_ids[e], s = seq[e];
  hop[p * MAX_LEN + s] = l2p[e];
  atomicMax(&plen[p], s + 1);
}

// ---------------------------------------------------------------------------
// Embedding MLP: 1 -> 16 -> 32, relu/relu (tiny; VALU)
// ---------------------------------------------------------------------------
__global__ void rn_embed(const float* __restrict__ x,
                         const float* __restrict__ w1, const float* __restrict__ b1,
                         const float* __restrict__ w2, const float* __restrict__ b2,
                         float* __restrict__ st, int n) {
  int i = blockIdx.x * blockDim.x + threadIdx.x;
  if (i >= n) return;
  float xi = x[i];
  float h[16];
#pragma unroll
  for (int j = 0; j < 16; ++j) h[j] = fmaxf(xi * w1[j] + b1[j], 0.0f);
#pragma unroll 4
  for (int o = 0; o < 32; ++o) {
    float s = b2[o];
#pragma unroll
    for (int j = 0; j < 16; ++j) s += h[j] * w2[j * 32 + o];
    st[(size_t)i * 32 + o] = fmaxf(s, 0.0f);
  }
}

// ---------------------------------------------------------------------------
// lgx[r][0..95] = state[r][0..31] @ Wk[32][96] + bias[0..95]
// One wave per 16-row tile. 48 WMMAs.
// ---------------------------------------------------------------------------
__global__ void rn_link_gx(const float* __restrict__ state,
                           const float* __restrict__ Wk,
                           const float* __restrict__ bias,
                           float* __restrict__ out) {
  int base = blockIdx.x * 16;
  int l = threadIdx.x, lo = l & 15, half = l >> 4;
  v8f acc[6];
#pragma unroll
  for (int nt = 0; nt < 6; ++nt) {
    float bv = bias[nt * 16 + lo];
#pragma unroll
    for (int v = 0; v < 8; ++v) acc[nt][v] = bv;
  }
#pragma unroll
  for (int kc = 0; kc < 8; ++kc) {
    v2f a;
    const float* ap = state + (size_t)(base + lo) * 32 + kc * 4 + 2 * half;
    a.x = ap[0]; a.y = ap[1];
#pragma unroll
    for (int nt = 0; nt < 6; ++nt) {
      v2f b;
      b.x = Wk[(kc * 4 + 2 * half) * GATE + nt * 16 + lo];
      b.y = Wk[(kc * 4 + 2 * half + 1) * GATE + nt * 16 + lo];
      acc[nt] = wmma4(a, b, acc[nt]);
    }
  }
#pragma unroll
  for (int nt = 0; nt < 6; ++nt)
#pragma unroll
    for (int v = 0; v < 8; ++v)
      out[(size_t)(base + v + 8 * half) * GATE + nt * 16 + lo] = acc[nt][v];
}

// ---------------------------------------------------------------------------
// Path GRU scan: one wave per 16 paths, 15 steps.
// gx gathered from per-link precomputed lgx; gh = h @ pg_rk via WMMA with
// pg_rk resident in registers (48 B-fragments).
// ---------------------------------------------------------------------------
__global__ void rn_path_scan(const float* __restrict__ lgx,
                             const float* __restrict__ Wrk,   // pg_rk [32][96]
                             const float* __restrict__ pg_b,  // [2][96]
                             const int*  __restrict__ hop,    // [n_paths][15]
                             const int*  __restrict__ plen,
                             float* __restrict__ path_state) {
  __shared__ float s_h[16 * 33];     // padded transpose buffer
  __shared__ int   s_hop[16 * MAX_LEN];
  __shared__ int   s_len[16];

  int base = blockIdx.x * 16;
  int l = threadIdx.x, lo = l & 15, half = l >> 4;

  for (int i = l; i < 16 * MAX_LEN; i += 32) s_hop[i] = hop[(size_t)base * MAX_LEN + i];
  if (l < 16) s_len[l] = plen[base + l];
  __syncthreads();

  // pg_rk B-fragments resident in registers
  v2f Brk[8][6];
#pragma unroll
  for (int kc = 0; kc < 8; ++kc)
#pragma unroll
    for (int nt = 0; nt < 6; ++nt) {
      Brk[kc][nt].x = Wrk[(kc * 4 + 2 * half) * GATE + nt * 16 + lo];
      Brk[kc][nt].y = Wrk[(kc * 4 + 2 * half + 1) * GATE + nt * 16 + lo];
    }
  float b1v[6];
#pragma unroll
  for (int nt = 0; nt < 6; ++nt) b1v[nt] = pg_b[GATE + nt * 16 + lo];

  int mlen[8];
#pragma unroll
  for (int v = 0; v < 8; ++v) mlen[v] = s_len[v + 8 * half];

  // h state in C/D layout: hD[j][v] = h[base + v + 8*half][j*16 + lo]
  v8f hD[2];
#pragma unroll
  for (int j = 0; j < 2; ++j)
#pragma unroll
    for (int v = 0; v < 8; ++v)
      hD[j][v] = path_state[(size_t)(base + v + 8 * half) * P_DIM + j * 16 + lo];

  for (int s = 0; s < MAX_LEN; ++s) {
    // D-layout -> LDS -> A-layout fragments of h
#pragma unroll
    for (int j = 0; j < 2; ++j)
#pragma unroll
      for (int v = 0; v < 8; ++v)
        s_h[(v + 8 * half) * 33 + j * 16 + lo] = hD[j][v];
    __syncthreads();
    v2f hA[8];
#pragma unroll
    for (int kc = 0; kc < 8; ++kc) {
      hA[kc].x = s_h[lo * 33 + kc * 4 + 2 * half];
      hA[kc].y = s_h[lo * 33 + kc * 4 + 2 * half + 1];
    }
    __syncthreads();

    // gather gx rows (D layout) from per-link precomputed gates
    v8f gx[6];
#pragma unroll
    for (int v = 0; v < 8; ++v) {
      int li = s_hop[(v + 8 * half) * MAX_LEN + s];
      if (li >= 0) {
        const float* src = lgx + (size_t)li * GATE;
#pragma unroll
        for (int nt = 0; nt < 6; ++nt) gx[nt][v] = src[nt * 16 + lo];
      } else {
#pragma unroll
        for (int nt = 0; nt < 6; ++nt) gx[nt][v] = 0.0f;
      }
    }

    // gh = h @ pg_rk + b[1]
    v8f gh[6];
#pragma unroll
    for (int nt = 0; nt < 6; ++nt)
#pragma unroll
      for (int v = 0; v < 8; ++v) gh[nt][v] = b1v[nt];
#pragma unroll
    for (int kc = 0; kc < 8; ++kc)
#pragma unroll
      for (int nt = 0; nt < 6; ++nt)
        gh[nt] = wmma4(hA[kc], Brk[kc][nt], gh[nt]);

    // GRU gates (z, r, hhat), masked update
#pragma unroll
    for (int j = 0; j < 2; ++j)
#pragma unroll
      for (int v = 0; v < 8; ++v) {
        float z = sigmoidf_(gx[j][v]     + gh[j][v]);
        float r = sigmoidf_(gx[2 + j][v] + gh[2 + j][v]);
        float c = tanhf(gx[4 + j][v] + r * gh[4 + j][v]);
        float hn = z * hD[j][v] + (1.0f - z) * c;
        hD[j][v] = (s < mlen[v]) ? hn : hD[j][v];
      }
  }

#pragma unroll
  for (int j = 0; j < 2; ++j)
#pragma unroll
    for (int v = 0; v < 8; ++v)
      path_state[(size_t)(base + v + 8 * half) * P_DIM + j * 16 + lo] = hD[j][v];
}

// ---------------------------------------------------------------------------
// Link-side aggregation: psum[link] += path_state[path] over incident edges
// ---------------------------------------------------------------------------
__global__ void rn_zero(float* p, int n) {
  int i = blockIdx.x * blockDim.x + threadIdx.x;
  if (i < n) p[i] = 0.0f;
}

__global__ void rn_scatter_add(const float* __restrict__ ps,
                               const int* __restrict__ p2l,   // path id per edge
                               const int* __restrict__ seql,  // link id per edge
                               float* __restrict__ psum, int E) {
  long i = (long)blockIdx.x * blockDim.x + threadIdx.x;
  if (i >= (long)E * 32) return;
  int e = (int)(i >> 5), d = (int)(i & 31);
  atomicAdd(&psum[(size_t)seql[e] * 32 + d], ps[(size_t)p2l[e] * 32 + d]);
}

// ---------------------------------------------------------------------------
// Link GRU: one wave per 16 links. gx = psum@lg_k+b0, gh = h@lg_rk+b1
// ---------------------------------------------------------------------------
__global__ void rn_link_gru(const float* __restrict__ psum,
                            const float* __restrict__ Wk,   // lg_k [32][96]
                            const float* __restrict__ Wrk,  // lg_rk [32][96]
                            const float* __restrict__ lg_b, // [2][96]
                            float* __restrict__ link_state) {
  int base = blockIdx.x * 16;
  int l = threadIdx.x, lo = l & 15, half = l >> 4;
  v8f gx[6], gh[6];
#pragma unroll
  for (int nt = 0; nt < 6; ++nt) {
    float b0 = lg_b[nt * 16 + lo], b1 = lg_b[GATE + nt * 16 + lo];
#pragma unroll
    for (int v = 0; v < 8; ++v) { gx[nt][v] = b0; gh[nt][v] = b1; }
  }
#pragma unroll
  for (int kc = 0; kc < 8; ++kc) {
    v2f ax, ah;
    const float* px = psum       + (size_t)(base + lo) * 32 + kc * 4 + 2 * half;
    const float* ph = link_state + (size_t)(base + lo) * 32 + kc * 4 + 2 * half;
    ax.x = px[0]; ax.y = px[1];
    ah.x = ph[0]; ah.y = ph[1];
#pragma unroll
    for (int nt = 0; nt < 6; ++nt) {
      v2f bx, bh;
      bx.x = Wk[(kc * 4 + 2 * half) * GATE + nt * 16 + lo];
      bx.y = Wk[(kc * 4 + 2 * half + 1) * GATE + nt * 16 + lo];
      bh.x = Wrk[(kc * 4 + 2 * half) * GATE + nt * 16 + lo];
      bh.y = Wrk[(kc * 4 + 2 * half + 1) * GATE + nt * 16 + lo];
      gx[nt] = wmma4(ax, bx, gx[nt]);
      gh[nt] = wmma4(ah, bh, gh[nt]);
    }
  }
  v8f hD[2];
#pragma unroll
  for (int j = 0; j < 2; ++j)
#pragma unroll
    for (int v = 0; v < 8; ++v)
      hD[j][v] = link_state[(size_t)(base + v + 8 * half) * 32 + j * 16 + lo];
#pragma unroll
  for (int j = 0; j < 2; ++j)
#pragma unroll
    for (int v = 0; v < 8; ++v) {
      float z = sigmoidf_(gx[j][v]     + gh[j][v]);
      float r = sigmoidf_(gx[2 + j][v] + gh[2 + j][v]);
      float c = tanhf(gx[4 + j][v] + r * gh[4 + j][v]);
      hD[j][v] = z * hD[j][v] + (1.0f - z) * c;
    }
#pragma unroll
  for (int j = 0; j < 2; ++j)
#pragma unroll
    for (int v = 0; v < 8; ++v)
      link_state[(size_t)(base + v + 8 * half) * 32 + j * 16 + lo] = hD[j][v];
}

// ---------------------------------------------------------------------------
// Readout: 32 -> 256 -> 256 -> 1, one wave per 16 paths, all WMMA.
// ---------------------------------------------------------------------------
__global__ void rn_readout(const float* __restrict__ ps,
                           const float* __restrict__ w1, const float* __restrict__ b1,
                           const float* __restrict__ w2, const float* __restrict__ b2,
                           const float* __restrict__ w3, const float* __restrict__ b3,
                           float* __restrict__ out) {
  __shared__ float s_h[16 * 257];   // padded [16][256]
  int base = blockIdx.x * 16;
  int l = threadIdx.x, lo = l & 15, half = l >> 4;

  v8f acc[16];
  // layer 1: [16,32] @ [32,256]
#pragma unroll
  for (int nt = 0; nt < 16; ++nt) {
    float bv = b1[nt * 16 + lo];
#pragma unroll
    for (int v = 0; v < 8; ++v) acc[nt][v] = bv;
  }
#pragma unroll
  for (int kc = 0; kc < 8; ++kc) {
    v2f a;
    const float* ap = ps + (size_t)(base + lo) * 32 + kc * 4 + 2 * half;
    a.x = ap[0]; a.y = ap[1];
#pragma unroll
    for (int nt = 0; nt < 16; ++nt) {
      v2f b;
      b.x = w1[(kc * 4 + 2 * half) * RO + nt * 16 + lo];
      b.y = w1[(kc * 4 + 2 * half + 1) * RO + nt * 16 + lo];
      acc[nt] = wmma4(a, b, acc[nt]);
    }
  }
#pragma unroll
  for (int nt = 0; nt < 16; ++nt)
#pragma unroll
    for (int v = 0; v < 8; ++v)
      s_h[(v + 8 * half) * 257 + nt * 16 + lo] = fmaxf(acc[nt][v], 0.0f);
  __syncthreads();

  // layer 2: [16,256] @ [256,256]
#pragma unroll
  for (int nt = 0; nt < 16; ++nt) {
    float bv = b2[nt * 16 + lo];
#pragma unroll
    for (int v = 0; v < 8; ++v) acc[nt][v] = bv;
  }
  for (int kc = 0; kc < 64; ++kc) {
    v2f a;
    a.x = s_h[lo * 257 + kc * 4 + 2 * half];
    a.y = s_h[lo * 257 + kc * 4 + 2 * half + 1];
#pragma unroll
    for (int nt = 0; nt < 16; ++nt) {
      v2f b;
      b.x = w2[(kc * 4 + 2 * half) * RO + nt * 16 + lo];
      b.y = w2[(kc * 4 + 2 * half + 1) * RO + nt * 16 + lo];
      acc[nt] = wmma4(a, b, acc[nt]);
    }
  }
  __syncthreads();
#pragma unroll
  for (int nt = 0; nt < 16; ++nt)
#pragma unroll
    for (int v = 0; v < 8; ++v)
      s_h[(v + 8 * half) * 257 + nt * 16 + lo] = fmaxf(acc[nt][v], 0.0f);
  __syncthreads();

  // layer 3: dot-256 per path (lanes 0..15, one path each)
  if (l < 16) {
    float s = b3[0];
    for (int n = 0; n < RO; ++n) s += s_h[l * 257 + n] * w3[n];
    out[base + l] = s;
  }
}

// ---------------------------------------------------------------------------
// Launch
// ---------------------------------------------------------------------------
extern "C" void kernel_launch(void* const* d_in, const int* in_sizes, int n_in,
                              void* d_out, int out_size, void* d_ws, size_t ws_size,
                              hipStream_t stream) {
  const float* traffic        = (const float*)d_in[0];
  const float* capacity       = (const float*)d_in[1];
  const int*   link_to_path   = (const int*)d_in[2];
  const int*   path_ids       = (const int*)d_in[3];
  const int*   sequence_path  = (const int*)d_in[4];
  const int*   path_to_link   = (const int*)d_in[5];
  const int*   sequence_links = (const int*)d_in[6];
  // d_in[7]=n_links, d_in[8]=n_paths (device scalars; use in_sizes instead)
  const float* le_w1 = (const float*)d_in[9];
  const float* le_b1 = (const float*)d_in[10];
  const float* le_w2 = (const float*)d_in[11];
  const float* le_b2 = (const float*)d_in[12];
  const float* pe_w1 = (const float*)d_in[13];
  const float* pe_b1 = (const float*)d_in[14];
  const float* pe_w2 = (const float*)d_in[15];
  const float* pe_b2 = (const float*)d_in[16];
  const float* pg_k  = (const float*)d_in[17];
  const float* pg_rk = (const float*)d_in[18];
  const float* pg_b  = (const float*)d_in[19];
  const float* lg_k  = (const float*)d_in[20];
  const float* lg_rk = (const float*)d_in[21];
  const float* lg_b  = (const float*)d_in[22];
  const float* ro_w1 = (const float*)d_in[23];
  const float* ro_b1 = (const float*)d_in[24];
  const float* ro_w2 = (const float*)d_in[25];
  const float* ro_b2 = (const float*)d_in[26];
  const float* ro_w3 = (const float*)d_in[27];
  const float* ro_b3 = (const float*)d_in[28];

  const int n_paths = in_sizes[0];
  const int n_links = in_sizes[1];
  const int E       = in_sizes[2];

  // workspace carve-out (256B aligned)
  char* w = (char*)d_ws;
  auto carve = [&](size_t bytes) {
    void* p = (void*)w;
    w += (bytes + 255) & ~(size_t)255;
    return p;
  };
  float* link_state = (float*)carve((size_t)n_links * 32 * 4);
  float* path_state = (float*)carve((size_t)n_paths * 32 * 4);
  float* lgx        = (float*)carve((size_t)n_links * GATE * 4);
  float* psum       = (float*)carve((size_t)n_links * 32 * 4);
  int*   hop        = (int*)carve((size_t)n_paths * MAX_LEN * 4);
  int*   plen       = (int*)carve((size_t)n_paths * 4);

  const int TB = 256;
  // hop table
  rn_init_hops<<<(n_paths * MAX_LEN + TB - 1) / TB, TB, 0, stream>>>(hop, plen, n_paths);
  rn_fill_hops<<<(E + TB - 1) / TB, TB, 0, stream>>>(path_ids, sequence_path,
                                                     link_to_path, hop, plen, E);
  // embeddings
  rn_embed<<<(n_links + TB - 1) / TB, TB, 0, stream>>>(capacity, le_w1, le_b1, le_w2, le_b2,
                                                       link_state, n_links);
  rn_embed<<<(n_paths + TB - 1) / TB, TB, 0, stream>>>(traffic, pe_w1, pe_b1, pe_w2, pe_b2,
                                                       path_state, n_paths);

  const int link_tiles = n_links / 16;   // 625
  const int path_tiles = n_paths / 16;   // 7500

  for (int t = 0; t < T_ITERS; ++t) {
    rn_link_gx<<<link_tiles, 32, 0, stream>>>(link_state, pg_k, pg_b, lgx);
    rn_path_scan<<<path_tiles, 32, 0, stream>>>(lgx, pg_rk, pg_b, hop, plen, path_state);
    rn_zero<<<(n_links * 32 + TB - 1) / TB, TB, 0, stream>>>(psum, n_links * 32);
    long sc_threads = (long)E * 32;
    rn_scatter_add<<<(unsigned)((sc_threads + TB - 1) / TB), TB, 0, stream>>>(
        path_state, path_to_link, sequence_links, psum, E);
    rn_link_gru<<<link_tiles, 32, 0, stream>>>(psum, lg_k, lg_rk, lg_b, link_state);
  }

  rn_readout<<<path_tiles, 32, 0, stream>>>(path_state, ro_w1, ro_b1, ro_w2, ro_b2,
                                            ro_w3, ro_b3, (float*)d_out);
}